// TreePUInfoNCE_51539607552059
// MI455X (gfx1250) — compile-verified
//
#include <hip/hip_runtime.h>
#include <hip/hip_bf16.h>

// Problem constants (fixed by the reference).
#define B_SZ 2048
#define M_SZ 4096
#define D_SZ 512
#define MAXP 16

constexpr float TAU      = 0.07f;
constexpr float EPS_CE   = 0.02f;
constexpr float SIGMA    = 0.7f;

typedef float v2f __attribute__((ext_vector_type(2)));
typedef float v8f __attribute__((ext_vector_type(8)));

// ---------- helpers ----------
__device__ __forceinline__ unsigned int fkey(float v) {
  unsigned int u = __float_as_uint(v);
  return (u >> 31) ? ~u : (u | 0x80000000u);
}
__device__ __forceinline__ float funkey(unsigned int k) {
  unsigned int u = (k >> 31) ? (k & 0x7fffffffu) : ~k;
  return __uint_as_float(u);
}

// ---------- init ----------
__global__ void k_zero_u32(unsigned int* __restrict__ p, int n) {
  int i = blockIdx.x * blockDim.x + threadIdx.x;
  if (i < n) p[i] = 0u;
}

// ---------- L2 normalize rows; also store squared norm of the normalized row ----------
__global__ void k_normalize(const float* __restrict__ za, const float* __restrict__ zc,
                            float* __restrict__ za_n, float* __restrict__ zc_n,
                            float* __restrict__ na2, float* __restrict__ nc2) {
  int row = blockIdx.x;
  int tid = threadIdx.x;  // 256 threads, D=512 -> 2 elems/thread
  const float* src; float* dst; float* n2p;
  if (row < B_SZ) {
    src = za + (size_t)row * D_SZ; dst = za_n + (size_t)row * D_SZ; n2p = na2 + row;
  } else {
    int r = row - B_SZ;
    src = zc + (size_t)r * D_SZ;  dst = zc_n + (size_t)r * D_SZ;  n2p = nc2 + r;
  }
  float x0 = src[tid], x1 = src[tid + 256];
  __shared__ float red[256];
  __shared__ float s_inv, s_n2;
  red[tid] = x0 * x0 + x1 * x1;
  __syncthreads();
  for (int off = 128; off > 0; off >>= 1) {
    if (tid < off) red[tid] += red[tid + off];
    __syncthreads();
  }
  if (tid == 0) {
    float ss = red[0];
    float inv = 1.0f / fmaxf(sqrtf(ss), 1e-12f);
    s_inv = inv;
    s_n2 = ss * inv * inv;  // ||normalized row||^2
  }
  __syncthreads();
  dst[tid] = x0 * s_inv;
  dst[tid + 256] = x1 * s_inv;
  if (tid == 0) *n2p = s_n2;
}

// ---------- S = ZaN * ZcN^T via V_WMMA_F32_16X16X4_F32 ----------
// Each wave computes a 32x32 output tile as 2x2 fragments of 16x16.
// A-frag layout (16x4 f32): lane l holds A[l%16][k0 + 2*(l/16) + {0,1}] in {v0,v1}.
// B-frag layout (4x16 f32): lane l holds B[k0 + 2*(l/16) + {0,1}][l%16]; since we
// compute A*B^T with B rows = zc rows, B-frag element [k][n] = ZcN[n][k] -> same
// float2 load pattern as A, from ZcN.
__global__ void k_gemm_wmma(const float* __restrict__ A, const float* __restrict__ Bm,
                            float* __restrict__ S) {
  const int TN = M_SZ / 32;                   // 128 tiles in N
  int wid  = threadIdx.x >> 5;                // 4 waves / block
  int lane = threadIdx.x & 31;
  int w  = blockIdx.x * 4 + wid;              // 0 .. 8191
  int tm = w / TN;
  int tn = w - tm * TN;
  int mr = lane & 15;
  int kg = (lane >> 4) << 1;                  // 0 or 2
  const float* a0p = A  + (size_t)(tm * 32 + mr) * D_SZ + kg;
  const float* a1p = a0p + (size_t)16 * D_SZ;
  const float* b0p = Bm + (size_t)(tn * 32 + mr) * D_SZ + kg;
  const float* b1p = b0p + (size_t)16 * D_SZ;

  v8f c00 = {}; v8f c01 = {}; v8f c10 = {}; v8f c11 = {};
  for (int k = 0; k < D_SZ; k += 4) {
    v2f a0 = *(const v2f*)(a0p + k);
    v2f a1 = *(const v2f*)(a1p + k);
    v2f b0 = *(const v2f*)(b0p + k);
    v2f b1 = *(const v2f*)(b1p + k);
    c00 = __builtin_amdgcn_wmma_f32_16x16x4_f32(false, a0, false, b0, (short)0, c00, false, false);
    c01 = __builtin_amdgcn_wmma_f32_16x16x4_f32(false, a0, false, b1, (short)0, c01, false, false);
    c10 = __builtin_amdgcn_wmma_f32_16x16x4_f32(false, a1, false, b0, (short)0, c10, false, false);
    c11 = __builtin_amdgcn_wmma_f32_16x16x4_f32(false, a1, false, b1, (short)0, c11, false, false);
  }
  // C/D layout: VGPR r, lane l -> row = r + 8*(l>>4), col = l&15
  int rh  = (lane >> 4) << 3;
  int col = lane & 15;
  int n0  = tn * 32 + col;
  for (int r = 0; r < 8; ++r) {
    int m0 = tm * 32 + rh + r;
    int m1 = m0 + 16;
    S[(size_t)m0 * M_SZ + n0]      = c00[r];
    S[(size_t)m0 * M_SZ + n0 + 16] = c01[r];
    S[(size_t)m1 * M_SZ + n0]      = c10[r];
    S[(size_t)m1 * M_SZ + n0 + 16] = c11[r];
  }
}

// ---------- median pass 1: histogram of hi-16 of monotone key, off-diag only ----------
__global__ void k_hist1(const float* __restrict__ td, unsigned int* __restrict__ hist) {
  size_t stride = (size_t)gridDim.x * blockDim.x;
  size_t total  = (size_t)M_SZ * M_SZ;
  for (size_t i = (size_t)blockIdx.x * blockDim.x + threadIdx.x; i < total; i += stride) {
    int row = (int)(i >> 12);           // M = 4096 = 2^12
    int col = (int)(i & (M_SZ - 1));
    if (row == col) continue;
    unsigned int k = fkey(td[i]);
    atomicAdd(&hist[k >> 16], 1u);
  }
}

// ---------- locate hi-16 bins holding the two middle order statistics ----------
__global__ void k_scan1(const unsigned int* __restrict__ hist, unsigned int* __restrict__ sel) {
  __shared__ unsigned int csum[256];
  int tid = threadIdx.x;
  unsigned int s = 0;
  for (int i = 0; i < 256; ++i) s += hist[tid * 256 + i];
  csum[tid] = s;
  __syncthreads();
  if (tid == 0) {
    const unsigned long long N = (unsigned long long)M_SZ * M_SZ - M_SZ;  // even
    unsigned long long ranks[2] = { (N - 1) / 2, (N - 1) / 2 + 1 };
    for (int ri = 0; ri < 2; ++ri) {
      unsigned long long r = ranks[ri];
      unsigned long long cum = 0;
      int c = 0;
      while (c < 255 && cum + csum[c] <= r) { cum += csum[c]; ++c; }
      int bin = c * 256;
      while (bin < 65535 && cum + hist[bin] <= r) { cum += hist[bin]; ++bin; }
      sel[ri]     = (unsigned int)bin;  // hi16 value
      sel[2 + ri] = (unsigned int)cum;  // exclusive prefix below bin
    }
  }
}

// ---------- median pass 2: lo-16 histograms within the selected hi bins ----------
__global__ void k_hist2(const float* __restrict__ td, const unsigned int* __restrict__ sel,
                        unsigned int* __restrict__ h2a, unsigned int* __restrict__ h2b) {
  unsigned int hiA = sel[0], hiB = sel[1];
  size_t stride = (size_t)gridDim.x * blockDim.x;
  size_t total  = (size_t)M_SZ * M_SZ;
  for (size_t i = (size_t)blockIdx.x * blockDim.x + threadIdx.x; i < total; i += stride) {
    int row = (int)(i >> 12);
    int col = (int)(i & (M_SZ - 1));
    if (row == col) continue;
    unsigned int k  = fkey(td[i]);
    unsigned int hi = k >> 16;
    if (hi == hiA) atomicAdd(&h2a[k & 0xFFFFu], 1u);
    if (hi == hiB) atomicAdd(&h2b[k & 0xFFFFu], 1u);
  }
}

// ---------- finish selection; write 0.5/(scale+1e-12) (folds symmetrization 0.5) ----------
__global__ void k_scan2(const unsigned int* __restrict__ sel,
                        const unsigned int* __restrict__ h2a,
                        const unsigned int* __restrict__ h2b,
                        float* __restrict__ scale_out) {
  if (threadIdx.x != 0 || blockIdx.x != 0) return;
  const unsigned long long N = (unsigned long long)M_SZ * M_SZ - M_SZ;
  unsigned long long ranks[2] = { (N - 1) / 2, (N - 1) / 2 + 1 };
  float vals[2];
  for (int ri = 0; ri < 2; ++ri) {
    const unsigned int* h = ri ? h2b : h2a;
    unsigned long long rem = ranks[ri] - (unsigned long long)sel[2 + ri];
    unsigned long long cum = 0;
    int lo = 0;
    while (lo < 65535 && cum + h[lo] <= rem) { cum += h[lo]; ++lo; }
    unsigned int key = (sel[ri] << 16) | (unsigned int)lo;
    vals[ri] = funkey(key);
  }
  float scale = 0.5f * (vals[0] + vals[1]);   // linear interp at q=0.5, even count
  scale_out[0] = 0.5f / (scale + 1e-12f);
}

// ---------- fused per-anchor-row loss ----------
__global__ void __launch_bounds__(256)
k_row_loss(const float* __restrict__ S, const unsigned char* __restrict__ mask,
           const float* __restrict__ td, const float* __restrict__ na2,
           const float* __restrict__ nc2, const float* __restrict__ scale,
           float* __restrict__ loss) {
  int b   = blockIdx.x;
  int tid = threadIdx.x;  // 256
  __shared__ float s_logit[M_SZ];      // 16 KB
  __shared__ float red[256];
  __shared__ int   s_pos[MAXP];
  __shared__ int   s_cnt;
  __shared__ float s_max, s_lse, s_pospart, s_wsum;

  if (tid == 0) s_cnt = 0;
  __syncthreads();

  const unsigned char* mrow = mask + (size_t)b * M_SZ;
  // collect positive indices (<= N_POS = 4 per row)
  for (int j = 0; j < 16; ++j) {
    int m = j * 256 + tid;
    if (mrow[m]) {
      int o = atomicAdd(&s_cnt, 1);
      if (o < MAXP) s_pos[o] = m;
    }
  }

  // logits + row max
  float a2 = na2[b];
  const float* srow = S + (size_t)b * M_SZ;
  float lmax = -3.4e38f;
  for (int j = 0; j < 16; ++j) {
    int m = j * 256 + tid;
    float sq = fmaxf(a2 + nc2[m] - 2.0f * srow[m], 0.0f);
    float lg = (-sq * 0.25f) / TAU;
    s_logit[m] = lg;
    lmax = fmaxf(lmax, lg);
  }
  red[tid] = lmax;
  __syncthreads();  // also publishes s_pos / s_cnt
  for (int off = 128; off > 0; off >>= 1) {
    if (tid < off) red[tid] = fmaxf(red[tid], red[tid + off]);
    __syncthreads();
  }
  if (tid == 0) s_max = red[0];
  __syncthreads();

  // log-sum-exp
  float mx = s_max;
  float es = 0.0f;
  for (int j = 0; j < 16; ++j) es += expf(s_logit[j * 256 + tid] - mx);
  red[tid] = es;
  __syncthreads();
  for (int off = 128; off > 0; off >>= 1) {
    if (tid < off) red[tid] += red[tid + off];
    __syncthreads();
  }
  if (tid == 0) {
    float lse = mx + logf(red[0]);
    s_lse = lse;
    int c = s_cnt < MAXP ? s_cnt : MAXP;
    float pp = 0.0f;
    for (int p = 0; p < c; ++p) pp += s_logit[s_pos[p]] - lse;
    s_pospart = pp;
  }
  __syncthreads();

  // tree-smoothed negative weights
  float lse  = s_lse;
  int   cnt  = s_cnt < MAXP ? s_cnt : MAXP;
  float invc = 1.0f / (float)(cnt > 0 ? cnt : 1);
  float hInv = scale[0];                 // 0.5/(scale+1e-12)
  float wsum = 0.0f, wlp = 0.0f;
  for (int j = 0; j < 16; ++j) {
    int m = j * 256 + tid;
    float dt = 0.0f;
    for (int p = 0; p < cnt; ++p) {
      int pi = s_pos[p];
      dt += td[(size_t)pi * M_SZ + m] + td[(size_t)m * M_SZ + pi];
    }
    dt *= hInv * invc;                   // avg normalized symmetric tree dist
    float pf = mrow[m] ? 1.0f : 0.0f;
    float w = expf(-dt / SIGMA) * (1.0f - pf);
    wsum += w;
    wlp  += w * (s_logit[m] - lse);
  }
  red[tid] = wsum;
  __syncthreads();
  for (int off = 128; off > 0; off >>= 1) {
    if (tid < off) red[tid] += red[tid + off];
    __syncthreads();
  }
  if (tid == 0) s_wsum = red[0];
  __syncthreads();
  red[tid] = wlp;
  __syncthreads();
  for (int off = 128; off > 0; off >>= 1) {
    if (tid < off) red[tid] += red[tid + off];
    __syncthreads();
  }
  if (tid == 0) {
    float l = -((1.0f - EPS_CE) * invc * s_pospart +
                EPS_CE * red[0] / fmaxf(s_wsum, 1e-12f));
    loss[b] = l;
  }
}

// ---------- deterministic mean over B rows ----------
__global__ void k_mean(const float* __restrict__ loss, float* __restrict__ out) {
  __shared__ float red[256];
  int tid = threadIdx.x;
  float s = 0.0f;
  for (int j = 0; j < B_SZ / 256; ++j) s += loss[j * 256 + tid];
  red[tid] = s;
  __syncthreads();
  for (int off = 128; off > 0; off >>= 1) {
    if (tid < off) red[tid] += red[tid + off];
    __syncthreads();
  }
  if (tid == 0) out[0] = red[0] * (1.0f / (float)B_SZ);
}

// ---------- host ----------
extern "C" void kernel_launch(void* const* d_in, const int* in_sizes, int n_in,
                              void* d_out, int out_size, void* d_ws, size_t ws_size,
                              hipStream_t stream) {
  (void)in_sizes; (void)n_in; (void)out_size; (void)ws_size;
  const float*         za   = (const float*)d_in[0];          // [B,D]
  const float*         zc   = (const float*)d_in[1];          // [M,D]
  const unsigned char* mask = (const unsigned char*)d_in[2];  // [B,M] bool
  const float*         td   = (const float*)d_in[3];          // [M,M]
  float*               out  = (float*)d_out;

  char* ws = (char*)d_ws;
  float*        za_n  = (float*)(ws + 0);                     //  4 MB
  float*        zc_n  = (float*)(ws + 4194304);               //  8 MB
  float*        S     = (float*)(ws + 12582912);              // 32 MB
  unsigned int* hist1 = (unsigned int*)(ws + 46137344);       // 256 KB
  unsigned int* h2a   = (unsigned int*)(ws + 46399488);       // 256 KB
  unsigned int* h2b   = (unsigned int*)(ws + 46661632);       // 256 KB
  unsigned int* sel   = (unsigned int*)(ws + 46923776);       // 4 u32
  float*        scl   = (float*)(ws + 46924032);              // 1 f32
  float*        loss  = (float*)(ws + 46924288);              // 8 KB
  float*        na2   = (float*)(ws + 46932480);              // 8 KB
  float*        nc2   = (float*)(ws + 46940672);              // 16 KB

  // hist1/h2a/h2b are contiguous: zero all 3*65536 counters every call.
  k_zero_u32<<<(3 * 65536 + 255) / 256, 256, 0, stream>>>(hist1, 3 * 65536);
  k_normalize<<<B_SZ + M_SZ, 256, 0, stream>>>(za, zc, za_n, zc_n, na2, nc2);
  k_gemm_wmma<<<(B_SZ / 32) * (M_SZ / 32) / 4, 128, 0, stream>>>(za_n, zc_n, S);
  k_hist1<<<2048, 256, 0, stream>>>(td, hist1);
  k_scan1<<<1, 256, 0, stream>>>(hist1, sel);
  k_hist2<<<2048, 256, 0, stream>>>(td, sel, h2a, h2b);
  k_scan2<<<1, 32, 0, stream>>>(sel, h2a, h2b, scl);
  k_row_loss<<<B_SZ, 256, 0, stream>>>(S, mask, td, na2, nc2, scl, loss);
  k_mean<<<1, 256, 0, stream>>>(loss, out);
}